// RadonBackward_16947940950668
// MI455X (gfx1250) — compile-verified
//
#include <hip/hip_runtime.h>
#include <cstdint>

#define N_B   16
#define N_A   360
#define N_P   512
#define IMG   512
#define TPB   256

#define BUF_FLOATS (N_B * N_P)          // 8192 floats = 32 KB per buffer
#define BUF_BYTES  (BUF_FLOATS * 4u)

typedef unsigned int v4u __attribute__((ext_vector_type(4)));
typedef int          v8i __attribute__((ext_vector_type(8)));
typedef int          v4i __attribute__((ext_vector_type(4)));

// Issue one TDM load: 2D tile of 16 rows x 2048 bytes (the 16 batch rows of
// angle `a`) from global into LDS at byte offset `lds_byte_addr`.
// D# per CDNA5 ISA ch.8: group0 = {count/lds/global/type}, group1 = dims.
// Groups 2/3 (and the trailing extended group) are zero: 2-D tensor.
__device__ __forceinline__ void tdm_stage(uint64_t gbase, int a,
                                          uint32_t lds_byte_addr) {
  const uint64_t ga = gbase + (uint64_t)a * (uint64_t)(N_P * 4); // row of angle a, batch 0

  v4u g0;
  g0.x = 1u;                               // count=1, is_restore=0, gather off
  g0.y = lds_byte_addr;                    // lds_addr  (bits 63:32)
  g0.z = (uint32_t)ga;                     // global_addr[31:0]
  g0.w = (uint32_t)(ga >> 32) | 0x80000000u; // global_addr[56:32] | type=2

  v8i g1;
  g1.s0 = 0x00020000;                      // workgroup_mask=0, data_size=2 (4B)
  g1.s1 = (N_P << 16);                     // tensor_dim0 = 512 (bits 79:48)
  g1.s2 = (N_B << 16);                     // tensor_dim1 = 16  (bits 111:80)
  g1.s3 = (N_P << 16);                     // tile_dim0   = 512 (bits 127:112)
  g1.s4 = N_B;                             // tile_dim1=16, tile_dim2=0
  g1.s5 = N_A * N_P;                       // tensor_dim0_stride = 184320 elems
  g1.s6 = 0;
  g1.s7 = 0;

  v4i g2 = {0, 0, 0, 0};                   // 2-D tensor: groups 2/3 unused
  v4i g3 = {0, 0, 0, 0};
  v8i g4 = {0, 0, 0, 0, 0, 0, 0, 0};       // extra group (clang-23 6-arg form)

  __builtin_amdgcn_tensor_load_to_lds(g0, g1, g2, g3, g4, 0);
}

__global__ __launch_bounds__(TPB) void radon_bp_kernel(
    const float* __restrict__ sino,       // [16,1,360,512]
    const float* __restrict__ thetas,     // [360]
    const float* __restrict__ positions,  // [512]
    float* __restrict__ out)              // [16,1,512,512]
{
  // Double-buffered sinogram stage (+4 floats pad for the j+1 tail overread,
  // which is always weight-masked to zero) and a cos/sin table.
  __shared__ float smem[2 * BUF_FLOATS + 4];
  __shared__ float ctab[N_A];
  __shared__ float stab[N_A];

  const int tid = threadIdx.x;

  // Build cos/sin table once per block (precise, off the hot path).
  for (int a = tid; a < N_A; a += TPB) {
    const float th = thetas[a];
    ctab[a] = cosf(th);
    stab[a] = sinf(th);
  }

  const float pos0  = positions[0];
  const float invdp = 1.0f / (positions[1] - positions[0]);

  const int pix = blockIdx.x * TPB + tid;    // 1024 blocks x 256 threads
  const int x   = pix & (IMG - 1);
  const int y   = pix >> 9;
  const float xf = (float)x - (float)(IMG - 1) * 0.5f;
  const float yf = (float)y - (float)(IMG - 1) * 0.5f;

  float acc[N_B];
#pragma unroll
  for (int b = 0; b < N_B; ++b) acc[b] = 0.0f;

  // Low 32 bits of a generic pointer to a __shared__ object are the LDS
  // byte offset (aperture bits live in the high half).
  const uint32_t smem_base = (uint32_t)(uintptr_t)(void*)smem;
  const uint64_t gbase     = (uint64_t)(uintptr_t)(const void*)sino;

  // Prologue: wave 0 DMAs angle 0 into buffer 0 via the Tensor Data Mover.
  // TDM ignores EXEC, so guard with a real branch (s_cbranch_execz skip).
  if (tid < 32) {
    tdm_stage(gbase, 0, smem_base);
  }

  for (int a = 0; a < N_A; ++a) {
    // Wave 0: kick off next angle's DMA into the other buffer, then wait
    // only for the current angle's transfer (TENSORcnt is in-order), so the
    // next copy overlaps this angle's compute on all 8 waves.
    if (tid < 32) {
      if (a + 1 < N_A) {
        tdm_stage(gbase, a + 1,
                  smem_base + (uint32_t)((a + 1) & 1) * BUF_BYTES);
        __builtin_amdgcn_s_wait_tensorcnt(1);
      } else {
        __builtin_amdgcn_s_wait_tensorcnt(0);
      }
    }
    __syncthreads();  // stage-a tile (and trig table) visible to all waves

    // Per-pixel index math, shared across all 16 batches.
    const float c = ctab[a];
    const float s = stab[a];
    const float t    = fmaf(xf, c, yf * s);
    const float fidx = (t - pos0) * invdp;
    const float fl   = floorf(fidx);
    const float w    = fidx - fl;
    const int   i0   = (int)fl;

    const float w0 = ((i0 >= 0) && (i0 < N_P))      ? (1.0f - w) : 0.0f;
    const float w1 = ((i0 >= -1) && (i0 < N_P - 1)) ? w          : 0.0f;
    const int   j  = (i0 < 0) ? 0 : ((i0 > N_P - 1) ? (N_P - 1) : i0);

    // Adjacent taps -> single ds_read2_b32 per batch.
    const float* rb = smem + (a & 1) * BUF_FLOATS + j;
#pragma unroll
    for (int b = 0; b < N_B; ++b) {
      const float lo = rb[b * N_P];
      const float hi = rb[b * N_P + 1];
      acc[b] = fmaf(w0, lo, fmaf(w1, hi, acc[b]));
    }
    __syncthreads();  // everyone done reading before this buffer is rewritten
  }

  // Coalesced writeback: out[b, 0, y, x].
#pragma unroll
  for (int b = 0; b < N_B; ++b) {
    out[(size_t)b * (IMG * IMG) + (size_t)pix] = acc[b];
  }
}

extern "C" void kernel_launch(void* const* d_in, const int* in_sizes, int n_in,
                              void* d_out, int out_size, void* d_ws, size_t ws_size,
                              hipStream_t stream) {
  (void)in_sizes; (void)n_in; (void)out_size; (void)d_ws; (void)ws_size;
  const float* sino      = (const float*)d_in[0];
  const float* thetas    = (const float*)d_in[1];
  const float* positions = (const float*)d_in[2];
  float* out             = (float*)d_out;

  dim3 grid((IMG * IMG) / TPB);   // 1024 workgroups
  dim3 block(TPB);                // 8 waves (wave32)
  radon_bp_kernel<<<grid, block, 0, stream>>>(sino, thetas, positions, out);
}